// Linear3D_76287209111586
// MI455X (gfx1250) — compile-verified
//
#include <hip/hip_runtime.h>

typedef __attribute__((ext_vector_type(2))) float v2f;
typedef __attribute__((ext_vector_type(8))) float v8f;

#define DDIM 512
#define DD   (512 * 512)
#define KC   64      // K chunk
#define LDA  68      // padded LDS row stride (floats): 4-float bank advance per row
#define NT   128     // j-tile per block
#define NCHUNK (DDIM / KC)

// CDNA5 async DMA: memory -> LDS, per-lane 16B, tracked by ASYNCcnt.
// VDST = LDS byte offset VGPR (low 32 bits of flat shared pointer = wave LDS
// address per ISA flat-aperture truncation), VADDR = 64-bit global address.
__device__ __forceinline__ void async_copy_b128(const float* gsrc, float* lds_dst) {
    uint32_t lds_off = (uint32_t)(uintptr_t)lds_dst;
    uint64_t gaddr   = (uint64_t)(uintptr_t)gsrc;
    asm volatile("global_load_async_to_lds_b128 %0, %1, off"
                 :: "v"(lds_off), "v"(gaddr)
                 : "memory");
}

__device__ __forceinline__ void wait_async_all() {
    asm volatile("s_wait_asynccnt 0x0" ::: "memory");
}

// Block: 256 threads = 8 waves (wave32). Each block: one i, 128(n) x 128(j) tile.
// Wave (wm in 0..3, wn in 0..1) computes a 32x64 sub-tile = 2x4 grid of 16x16 WMMA tiles.
__global__ __launch_bounds__(256) void linear3d_wmma_f32(
    const float* __restrict__ x,   // [128, 512, 512] (n, i, k)
    const float* __restrict__ w,   // [512, 512, 512] (i, j, k)
    const float* __restrict__ bias,// [512, 512]      (i, j)
    float* __restrict__ out)       // [128, 512, 512] (n, i, j)
{
    // Double-buffered chunks, filled by async-to-LDS DMA (no VGPR staging).
    __shared__ float As[2][128 * LDA];  // x chunk: 128 rows(n) x 64(k), padded
    __shared__ float Bs[2][NT  * LDA];  // w chunk: 128 rows(j) x 64(k), padded

    const int tid   = threadIdx.x;
    const int i     = blockIdx.y;
    const int jbase = blockIdx.x * NT;

    const float* xi = x + (size_t)i * DDIM;                        // + n*DD + k
    const float* wi = w + (size_t)i * DD + (size_t)jbase * DDIM;   // + j*D  + k

    const int wv   = tid >> 5;
    const int lane = tid & 31;
    const int lr   = lane & 15;   // N index within 16 / fragment row
    const int lh   = lane >> 4;   // lane half -> K/M sub-select
    const int m0   = (wv & 3) * 32;  // wave's n base (0,32,64,96)
    const int j0   = (wv >> 2) * 64; // wave's local j base (0 or 64)

    v8f acc[2][4];
    #pragma unroll
    for (int mi = 0; mi < 2; ++mi)
        #pragma unroll
        for (int ni = 0; ni < 4; ++ni)
            acc[mi][ni] = (v8f){0.f, 0.f, 0.f, 0.f, 0.f, 0.f, 0.f, 0.f};

    // Issue one chunk's A+B tiles as 16 async b128 DMAs per thread-block pass:
    // 256 threads x 8 iterations cover 128 rows x 16 float4 for each of A and B.
    auto issueChunk = [&](int c, int buf) {
        const int k0 = c * KC;
        #pragma unroll
        for (int p = 0; p < 8; ++p) {
            int idx = tid + p * 256;
            int row = idx >> 4;          // 0..127
            int c4  = (idx & 15) << 2;   // 0,4,...,60 (floats)
            async_copy_b128(xi + (size_t)row * DD   + k0 + c4, &As[buf][row * LDA + c4]);
            async_copy_b128(wi + (size_t)row * DDIM + k0 + c4, &Bs[buf][row * LDA + c4]);
        }
    };

    issueChunk(0, 0);
    wait_async_all();
    __syncthreads();

    for (int c = 0; c < NCHUNK; ++c) {
        const int buf = c & 1;
        // Prefetch next chunk into the other buffer; its last readers finished
        // before the barrier that ended iteration c-1, so writes are safe.
        if (c + 1 < NCHUNK) issueChunk(c + 1, buf ^ 1);

        #pragma unroll
        for (int kk = 0; kk < KC; kk += 4) {
            v2f a[2], bb[4];
            #pragma unroll
            for (int mi = 0; mi < 2; ++mi)
                a[mi] = *reinterpret_cast<const v2f*>(
                    &As[buf][(m0 + mi * 16 + lr) * LDA + kk + 2 * lh]);
            #pragma unroll
            for (int ni = 0; ni < 4; ++ni)
                bb[ni] = *reinterpret_cast<const v2f*>(
                    &Bs[buf][(j0 + ni * 16 + lr) * LDA + kk + 2 * lh]);
            #pragma unroll
            for (int mi = 0; mi < 2; ++mi)
                #pragma unroll
                for (int ni = 0; ni < 4; ++ni)
                    acc[mi][ni] = __builtin_amdgcn_wmma_f32_16x16x4_f32(
                        false, a[mi], false, bb[ni],
                        (short)0, acc[mi][ni], false, false);
        }

        if (c + 1 < NCHUNK) {
            wait_async_all();   // this wave's DMAs for chunk c+1 are in LDS
            __syncthreads();    // -> all waves' DMAs complete; readers of buf done
        }
    }

    // Epilogue: C/D layout -> lane lr = j-within-16, VGPR r -> n = 8*lh + r
    #pragma unroll
    for (int mi = 0; mi < 2; ++mi) {
        #pragma unroll
        for (int ni = 0; ni < 4; ++ni) {
            const int jg = jbase + j0 + ni * 16 + lr;
            const float bv = bias[(size_t)i * DDIM + jg];
            #pragma unroll
            for (int r = 0; r < 8; ++r) {
                const int n = m0 + mi * 16 + lh * 8 + r;
                out[(size_t)n * DD + (size_t)i * DDIM + jg] = acc[mi][ni][r] + bv;
            }
        }
    }
}

extern "C" void kernel_launch(void* const* d_in, const int* in_sizes, int n_in,
                              void* d_out, int out_size, void* d_ws, size_t ws_size,
                              hipStream_t stream) {
    (void)in_sizes; (void)n_in; (void)out_size; (void)d_ws; (void)ws_size;
    const float* x    = (const float*)d_in[0];
    const float* w    = (const float*)d_in[1];
    const float* bias = (const float*)d_in[2];
    float* out        = (float*)d_out;

    dim3 grid(DDIM / NT, DDIM);  // (4 j-tiles, 512 i)
    dim3 block(256);
    hipLaunchKernelGGL(linear3d_wmma_f32, grid, block, 0, stream, x, w, bias, out);
}